// GraphSAGELite_43465069036000
// MI455X (gfx1250) — compile-verified
//
#include <hip/hip_runtime.h>
#include <hip/hip_bf16.h>

typedef __attribute__((ext_vector_type(2))) float v2f;
typedef __attribute__((ext_vector_type(8))) float v8f;

#define CH 64            // IN_CH == HID == 64
#define LDS_PITCH 66     // 64 + 2 pad -> distinct LDS bank pairs per lane

// ---------------------------------------------------------------- utilities
__global__ void k_zero(float* __restrict__ p, size_t n) {
    size_t i = (size_t)blockIdx.x * blockDim.x + threadIdx.x;
    if (i < n) p[i] = 0.0f;
}

// in-degree: cnt[dst] += 1  (edge_index row 1)
__global__ void k_degree(const int* __restrict__ ei, float* __restrict__ cnt, int n_edges) {
    int e = blockIdx.x * blockDim.x + threadIdx.x;
    if (e < n_edges) atomicAdd(&cnt[ei[n_edges + e]], 1.0f);
}

// agg[dst,:] += x[src,:]  — 16 lanes per edge, float4 coalesced reads,
// global_atomic_add_f32 into L2 (randomly scattered dst rows).
__global__ void k_scatter(const float* __restrict__ x, const int* __restrict__ ei,
                          float* __restrict__ agg, int n_edges) {
    int t = blockIdx.x * blockDim.x + threadIdx.x;
    int e = t >> 4;
    if (e >= n_edges) return;
    int lane = t & 15;
    int src = ei[e];
    int dst = ei[n_edges + e];
    float4 v = ((const float4*)(x + (size_t)src * CH))[lane];
    float* a = agg + (size_t)dst * CH + lane * 4;
    atomicAdd(a + 0, v.x);
    atomicAdd(a + 1, v.y);
    atomicAdd(a + 2, v.z);
    atomicAdd(a + 3, v.w);
}

// ---------------------------------------------------------------- SAGE layer
// out[i,:] = relu( (agg[i,:] @ Wl^T) / max(cnt[i],1) + bl + x[i,:] @ Wr^T )
// One wave per 16-node tile; 4 accumulators of 16x16 per matmul; K stepped by 4
// with V_WMMA_F32_16X16X4_F32 (full fp32 precision on the matrix pipe).
__global__ __launch_bounds__(256) void k_sage_gemm(
    const float* __restrict__ agg, const float* __restrict__ xin,
    const float* __restrict__ Wl,  const float* __restrict__ bl,
    const float* __restrict__ Wr,  const float* __restrict__ cnt,
    float* __restrict__ out, int n_nodes, int ntiles)
{
    __shared__ float sWl[CH * LDS_PITCH];
    __shared__ float sWr[CH * LDS_PITCH];
    // stage both 64x64 weight matrices into LDS (bank-conflict-free pitch)
    for (int i = threadIdx.x; i < CH * CH; i += 256) {
        int r = i >> 6, c = i & 63;
        sWl[r * LDS_PITCH + c] = Wl[i];
        sWr[r * LDS_PITCH + c] = Wr[i];
    }
    __syncthreads();

    int wv   = threadIdx.x >> 5;
    int tile = blockIdx.x * 8 + wv;
    if (tile >= ntiles) return;           // wave-uniform exit (EXEC stays all-1)

    int lane = threadIdx.x & 31;
    int half = lane >> 4;                 // 0: K pair {0,1}, 1: K pair {2,3}
    int l16  = lane & 15;

    v8f accL[4] = {};                     // agg @ Wl^T
    v8f accR[4] = {};                     // x   @ Wr^T

    // A fragment base: row m = l16 of this tile, columns kb + 2*half + {0,1}
    int arow = tile * 16 + l16;
    if (arow > n_nodes - 1) arow = n_nodes - 1;     // clamp (tail tiles)
    const float* aBase = agg + (size_t)arow * CH + 2 * half;
    const float* xBase = xin + (size_t)arow * CH + 2 * half;

#pragma unroll
    for (int kb = 0; kb < CH; kb += 4) {
        v2f aL = *(const v2f*)(aBase + kb);
        v2f aR = *(const v2f*)(xBase + kb);
#pragma unroll
        for (int nt = 0; nt < 4; ++nt) {
            // B fragment: B[k][n] = W[n][k], n = nt*16 + l16, k = kb + 2*half + {0,1}
            const float* pl = sWl + (nt * 16 + l16) * LDS_PITCH + kb + 2 * half;
            const float* pr = sWr + (nt * 16 + l16) * LDS_PITCH + kb + 2 * half;
            v2f bL = *(const v2f*)pl;
            v2f bR = *(const v2f*)pr;
            accL[nt] = __builtin_amdgcn_wmma_f32_16x16x4_f32(
                false, aL, false, bL, (short)0, accL[nt], false, false);
            accR[nt] = __builtin_amdgcn_wmma_f32_16x16x4_f32(
                false, aR, false, bR, (short)0, accR[nt], false, false);
        }
    }

    // epilogue: per-row mean scaling, bias, residual lin_r, relu
    float invc[8];
#pragma unroll
    for (int r = 0; r < 8; ++r) {
        int node = tile * 16 + r + 8 * half;
        int cidx = node < n_nodes ? node : n_nodes - 1;
        invc[r] = 1.0f / fmaxf(cnt[cidx], 1.0f);
    }
#pragma unroll
    for (int nt = 0; nt < 4; ++nt) {
        float bias = bl[nt * 16 + l16];
#pragma unroll
        for (int r = 0; r < 8; ++r) {
            int node = tile * 16 + r + 8 * half;
            if (node < n_nodes) {
                float v = accL[nt][r] * invc[r] + bias + accR[nt][r];
                out[(size_t)node * CH + nt * 16 + l16] = fmaxf(v, 0.0f);
            }
        }
    }
}

// ---------------------------------------------------------------- pooling
__global__ void k_gcnt(const int* __restrict__ batch, float* __restrict__ gcnt, int n_nodes) {
    int i = blockIdx.x * blockDim.x + threadIdx.x;
    if (i < n_nodes) atomicAdd(&gcnt[batch[i]], 1.0f);
}

__global__ void k_pool(const float* __restrict__ h, const int* __restrict__ batch,
                       float* __restrict__ gsum, int n_nodes) {
    int t = blockIdx.x * blockDim.x + threadIdx.x;
    int node = t >> 4;
    if (node >= n_nodes) return;
    int lane = t & 15;
    int g = batch[node];
    float4 v = ((const float4*)(h + (size_t)node * CH))[lane];
    float* p = gsum + (size_t)g * CH + lane * 4;
    atomicAdd(p + 0, v.x);
    atomicAdd(p + 1, v.y);
    atomicAdd(p + 2, v.z);
    atomicAdd(p + 3, v.w);
}

// out[g,o] = (gsum[g,:] @ Wc[o,:]) / max(gcnt,1) + bc[o]   (256x8, trivial)
__global__ void k_classifier(const float* __restrict__ gsum, const float* __restrict__ gcnt,
                             const float* __restrict__ Wc, const float* __restrict__ bc,
                             float* __restrict__ outp, int n_graphs) {
    int t = blockIdx.x * blockDim.x + threadIdx.x;
    if (t >= n_graphs * 8) return;
    int g = t >> 3, o = t & 7;
    float inv = 1.0f / fmaxf(gcnt[g], 1.0f);
    float s = 0.0f;
#pragma unroll 8
    for (int k = 0; k < CH; ++k) s += gsum[g * CH + k] * Wc[o * CH + k];
    outp[t] = s * inv + bc[o];
}

// ---------------------------------------------------------------- launcher
extern "C" void kernel_launch(void* const* d_in, const int* in_sizes, int n_in,
                              void* d_out, int out_size, void* d_ws, size_t ws_size,
                              hipStream_t stream) {
    const float* x     = (const float*)d_in[0];
    const int*   ei    = (const int*)d_in[1];   // [2, E] int32 (jax x64 disabled)
    const int*   batch = (const int*)d_in[2];
    const float* W1l   = (const float*)d_in[3];
    const float* b1l   = (const float*)d_in[4];
    const float* W1r   = (const float*)d_in[5];
    const float* W2l   = (const float*)d_in[6];
    const float* b2l   = (const float*)d_in[7];
    const float* W2r   = (const float*)d_in[8];
    const float* Wc    = (const float*)d_in[9];
    const float* bc    = (const float*)d_in[10];
    float* outp = (float*)d_out;

    const int n_nodes  = in_sizes[0] / CH;
    const int n_edges  = in_sizes[1] / 2;
    const int n_graphs = out_size / 8;
    const int ntiles   = (n_nodes + 15) / 16;

    // workspace carve-up (all fp32)
    float* ws   = (float*)d_ws;
    size_t NN   = (size_t)n_nodes;
    float* agg  = ws;                      // [N,64]
    float* h1   = agg  + NN * CH;          // [N,64]
    float* h2   = h1   + NN * CH;          // [N,64]
    float* cnt  = h2   + NN * CH;          // [N]
    float* gsum = cnt  + NN;               // [G,64]
    float* gcnt = gsum + (size_t)n_graphs * CH;  // [G]

    const int B = 256;
    size_t nAgg  = NN * CH;
    size_t nTail = NN + (size_t)n_graphs * CH + (size_t)n_graphs; // cnt+gsum+gcnt
    int scatBlocks = (int)(((size_t)n_edges * 16 + B - 1) / B);
    int poolBlocks = (int)((NN * 16 + B - 1) / B);
    int gemmBlocks = (ntiles + 7) / 8;

    // zero accumulator regions (ws is poisoned, and must be reset every call)
    k_zero<<<(int)((nAgg + B - 1) / B), B, 0, stream>>>(agg, nAgg);
    k_zero<<<(int)((nTail + B - 1) / B), B, 0, stream>>>(cnt, nTail);

    // in-degrees (shared by both layers)
    k_degree<<<(n_edges + B - 1) / B, B, 0, stream>>>(ei, cnt, n_edges);

    // ---- layer 1
    k_scatter<<<scatBlocks, B, 0, stream>>>(x, ei, agg, n_edges);
    k_sage_gemm<<<gemmBlocks, B, 0, stream>>>(agg, x, W1l, b1l, W1r, cnt, h1,
                                              n_nodes, ntiles);

    // ---- layer 2 (re-zero agg, then aggregate h1)
    k_zero<<<(int)((nAgg + B - 1) / B), B, 0, stream>>>(agg, nAgg);
    k_scatter<<<scatBlocks, B, 0, stream>>>(h1, ei, agg, n_edges);
    k_sage_gemm<<<gemmBlocks, B, 0, stream>>>(agg, h1, W2l, b2l, W2r, cnt, h2,
                                              n_nodes, ntiles);

    // ---- global mean pool + classifier
    k_gcnt<<<(n_nodes + B - 1) / B, B, 0, stream>>>(batch, gcnt, n_nodes);
    k_pool<<<poolBlocks, B, 0, stream>>>(h2, batch, gsum, n_nodes);
    k_classifier<<<(n_graphs * 8 + B - 1) / B, B, 0, stream>>>(gsum, gcnt, Wc, bc,
                                                               outp, n_graphs);
}